// GraphConvLayer_8091718385772
// MI455X (gfx1250) — compile-verified
//
#include <hip/hip_runtime.h>

typedef float v2f __attribute__((ext_vector_type(2)));
typedef float v8f __attribute__((ext_vector_type(8)));

#define TYPES 5
#define IN_F  8
#define OUT_F 8
#define BCOL  16

__global__ void rgcn_zero_kernel(float* __restrict__ p, int n) {
    int i = blockIdx.x * blockDim.x + threadIdx.x;
    if (i < n) p[i] = 0.0f;
}

// One wave (32 lanes) per edge-iteration.
// Per edge: D(16x16) = W_pad(16x8) @ x_src(8x16) via two V_WMMA_F32_16X16X4_F32,
// then 8 fp32 atomics per lane 0..15 into out[dst], bias fused.
__global__ void __launch_bounds__(256)
rgcn_edge_wmma(const float* __restrict__ x, const float* __restrict__ W,
               const float* __restrict__ Bv, const int* __restrict__ src,
               const int* __restrict__ dst, const int* __restrict__ et,
               float* __restrict__ out, int E) {
    // W pre-swizzled into the 16x4 A-matrix lane layout, two K-steps packed as float4:
    //   lane L (0..15): M=L, {K0,K1,K4,K5}; lane L+16: M=L, {K2,K3,K6,K7}; rows M>=8 zero.
    __shared__ float4 lds_a[TYPES * 32];
    __shared__ float  lds_bv[TYPES * OUT_F];

    const int tid = threadIdx.x;
    if (tid < TYPES * 32) {
        const int t = tid >> 5;
        const int l = tid & 31;
        const int m = l & 15;
        const int h = l >> 4;          // 0: K 0/1 (and 4/5), 1: K 2/3 (and 6/7)
        float4 v = make_float4(0.f, 0.f, 0.f, 0.f);
        if (m < OUT_F) {
            const float* w = W + (t * OUT_F + m) * IN_F;
            v.x = w[2 * h + 0];
            v.y = w[2 * h + 1];
            v.z = w[4 + 2 * h + 0];
            v.w = w[4 + 2 * h + 1];
        }
        lds_a[tid] = v;
    }
    if (tid < TYPES * OUT_F) lds_bv[tid] = Bv[tid];
    __syncthreads();

    const int lane = threadIdx.x & 31;
    const int col  = lane & 15;
    const int half = lane >> 4;
    const int wave   = (blockIdx.x * blockDim.x + threadIdx.x) >> 5;
    const int nwaves = (gridDim.x * blockDim.x) >> 5;

    for (int e = wave; e < E; e += nwaves) {
        // Edge scalars are wave-uniform: force to SGPRs so W/Bv addressing scalarizes.
        const int t = __builtin_amdgcn_readfirstlane(et[e]) - 1;   // 0..4
        const int s = __builtin_amdgcn_readfirstlane(src[e]);
        const int d = __builtin_amdgcn_readfirstlane(dst[e]);

        const int en = e + nwaves;                                 // prefetch next edge
        if (en < E) {
            __builtin_prefetch(&src[en], 0, 0);
            __builtin_prefetch(&dst[en], 0, 0);
            __builtin_prefetch(&et[en], 0, 0);
        }

        // A-matrix fragments from LDS (one b128 load).
        const float4 a4 = lds_a[t * 32 + lane];
        v2f a_lo = { a4.x, a4.y };   // K 0..3 step
        v2f a_hi = { a4.z, a4.w };   // K 4..7 step

        // B-matrix 4x16 per step: VGPR0 lanes0-15 = row(K) h*2, lanes16-31 = row h*2 (K+2);
        // each row of x is 64B -> coalesced across 16 lanes.
        const float* xp = x + ((size_t)s * IN_F + 2 * half) * BCOL + col;
        v2f b_lo = { xp[0],        xp[BCOL] };          // rows 0..3
        v2f b_hi = { xp[4 * BCOL], xp[5 * BCOL] };      // rows 4..7

        v8f c = { 0.f, 0.f, 0.f, 0.f, 0.f, 0.f, 0.f, 0.f };
        c = __builtin_amdgcn_wmma_f32_16x16x4_f32(false, a_lo, false, b_lo,
                                                  (short)0, c, false, false);
        c = __builtin_amdgcn_wmma_f32_16x16x4_f32(false, a_hi, false, b_hi,
                                                  (short)0, c, false, false);

        // D rows 0..7 = msg; scatter-add into out[dst] with bias fused.
        if (lane < 16) {
            float* op = out + (size_t)d * OUT_F * BCOL + col;
            #pragma unroll
            for (int r = 0; r < OUT_F; ++r) {
                const float bv = lds_bv[t * OUT_F + r];
                atomicAdd(op + r * BCOL, c[r] + bv);
            }
        }
    }
}

extern "C" void kernel_launch(void* const* d_in, const int* in_sizes, int n_in,
                              void* d_out, int out_size, void* d_ws, size_t ws_size,
                              hipStream_t stream) {
    const float* x  = (const float*)d_in[0];
    const float* W  = (const float*)d_in[1];
    const float* Bv = (const float*)d_in[2];
    const int* src  = (const int*)d_in[3];
    const int* dst  = (const int*)d_in[4];
    const int* et   = (const int*)d_in[5];
    float* out      = (float*)d_out;
    const int E = in_sizes[3];

    const int zt = 256;
    const int zb = (out_size + zt - 1) / zt;
    rgcn_zero_kernel<<<zb, zt, 0, stream>>>(out, out_size);

    const int threads = 256;                 // 8 waves/block
    const int blocks  = 2048;                // 16384 persistent waves (~100 edges each)
    rgcn_edge_wmma<<<blocks, threads, 0, stream>>>(x, W, Bv, src, dst, et, out, E);
}